// Model_72026601554419
// MI455X (gfx1250) — compile-verified
//
#include <hip/hip_runtime.h>
#include <hip/hip_bf16.h>

// Elementwise particle-dynamics step. B = 8,388,608 rows.
// HBM-bound (~503 MB @ 23.3 TB/s => ~22 us floor). One thread per row,
// wave32 lanes cover contiguous memory for the strided 5-float rows.
// Transcendentals use the CDNA5 hardware units (v_sin_f32 / v_cos_f32 /
// v_exp_f32) since the angle is pre-reduced to [-pi, pi).
//
// Inputs (setup_inputs order):
//   d_in[0] x                : float[B*5]
//   d_in[1] a                : float[B*2]
//   d_in[2] noise            : float[B*2]
//   d_in[3] pro_gains        : float[2]
//   d_in[4] pro_noise_ln_vars: float[2]
//   d_in[5] goal_radius      : float[1]
// Output: next_x float[B*5] followed by reached_target float[B] (0/1).

#define DT_STEP      0.1f
#define BOX_LIM      1.0f
#define PI_F         3.14159265358979323846f
#define TWO_PI_F     6.28318530717958647692f
#define INV_TWO_PI_F 0.15915494309189535f

__global__ __launch_bounds__(256) void particle_step_kernel(
    const float* __restrict__ x,
    const float* __restrict__ a,
    const float* __restrict__ noise,
    const float* __restrict__ pro_gains,
    const float* __restrict__ pro_noise_ln_vars,
    const float* __restrict__ goal_radius,
    float* __restrict__ out_next,    // B*5
    float* __restrict__ out_reached, // B
    int n)
{
    int i = blockIdx.x * blockDim.x + threadIdx.x;
    if (i >= n) return;

    // Uniform scalar parameters (uniform address -> scalar/L2 path).
    const float g0 = pro_gains[0];
    const float g1 = pro_gains[1];
    const float s0 = __expf(0.5f * pro_noise_ln_vars[0]); // sqrt(exp(v)) == exp(v/2)
    const float s1 = __expf(0.5f * pro_noise_ln_vars[1]);
    const float gr = goal_radius[0];

    // Prefetch a future tile of the streaming inputs (gfx1250 global_prefetch_b8).
    // Speculative: out-of-range prefetches are silently dropped.
    {
        const int pf = i + 8192;
        __builtin_prefetch(&x[(size_t)pf * 5], 0, 0);
        __builtin_prefetch(&a[(size_t)pf * 2], 0, 0);
        __builtin_prefetch(&noise[(size_t)pf * 2], 0, 0);
    }

    // Row loads. a/noise rows are 8B-aligned -> float2 (global_load_b64).
    const float px_in  = x[(size_t)i * 5 + 0];
    const float py_in  = x[(size_t)i * 5 + 1];
    const float ang_in = x[(size_t)i * 5 + 2];

    const float2 av = *reinterpret_cast<const float2*>(&a[(size_t)i * 2]);
    const float2 nv = *reinterpret_cast<const float2*>(&noise[(size_t)i * 2]);

    // w = std * noise ; vel / ang_vel
    const float vel     = fmaf(g0, av.x, s0 * nv.x);
    const float ang_vel = fmaf(g1, av.y, s1 * nv.y);

    // range_angle(ang + ang_vel*DT): Python-mod semantics via floor-division.
    //   r = mod(t + pi, 2pi) in [0, 2pi), then - pi.  Branch-free, 3 VALU ops.
    const float t   = fmaf(ang_vel, DT_STEP, ang_in) + PI_F;
    const float k   = floorf(t * INV_TWO_PI_F);
    const float ang = fmaf(-TWO_PI_F, k, t) - PI_F;

    // Angle is in [-pi, pi): hardware sin/cos fast path (v_sin_f32 / v_cos_f32).
    const float c = __cosf(ang);
    const float s = __sinf(ang);

    float px = fmaf(vel * c, DT_STEP, px_in);
    float py = fmaf(vel * s, DT_STEP, py_in);
    px = fminf(fmaxf(px, -BOX_LIM), BOX_LIM);
    py = fminf(fmaxf(py, -BOX_LIM), BOX_LIM);

    // next_x = [px, py, ang, vel, ang_vel]
    float* row = &out_next[(size_t)i * 5];
    row[0] = px;
    row[1] = py;
    row[2] = ang;
    row[3] = vel;
    row[4] = ang_vel;

    const float dist = sqrtf(fmaf(px, px, py * py));
    out_reached[i] = (dist <= gr) ? 1.0f : 0.0f;
}

extern "C" void kernel_launch(void* const* d_in, const int* in_sizes, int n_in,
                              void* d_out, int out_size, void* d_ws, size_t ws_size,
                              hipStream_t stream) {
    const float* x       = (const float*)d_in[0];
    const float* a       = (const float*)d_in[1];
    const float* noise   = (const float*)d_in[2];
    const float* gains   = (const float*)d_in[3];
    const float* lnvars  = (const float*)d_in[4];
    const float* gradius = (const float*)d_in[5];

    const int n = in_sizes[0] / 5;  // B rows

    float* out_next    = (float*)d_out;            // B*5 floats
    float* out_reached = out_next + (size_t)n * 5; // B floats

    const int block = 256;                         // 8 wave32 waves per block
    const int grid  = (n + block - 1) / block;

    particle_step_kernel<<<grid, block, 0, stream>>>(
        x, a, noise, gains, lnvars, gradius, out_next, out_reached, n);
}